// DirectMPIGO_1726576856423
// MI455X (gfx1250) — compile-verified
//
#include <hip/hip_runtime.h>
#include <math.h>

#define S_SAMPLES 256
#define CHUNK 32
#define NCHUNK (S_SAMPLES / CHUNK)      // 8
#define RAYS_PER_BLOCK 8                // 8 wave32's = 256 threads

// ---------- CDNA5 async global->LDS path (guarded; fallback = sync copy) ----
#if defined(__gfx1250__) && __has_builtin(__builtin_amdgcn_global_load_async_to_lds_b32) && __has_builtin(__builtin_amdgcn_s_wait_asynccnt)
#define USE_ASYNC_LDS 1
#else
#define USE_ASYNC_LDS 0
#endif

typedef __attribute__((address_space(1))) int gint_t;   // global (AS1) int
typedef __attribute__((address_space(3))) int lint_t;   // LDS (AS3) int

__device__ __forceinline__ void stage_b32(const float* g, float* l) {
#if USE_ASYNC_LDS
    // one b32 per lane; one instruction per wave -> ASYNCcnt += 1
    __builtin_amdgcn_global_load_async_to_lds_b32(
        (gint_t*)g, (lint_t*)l, /*offset=*/0, /*cpol=*/0);
#else
    *l = *g;
#endif
}

__device__ __forceinline__ void wait_async_le1() {
#if USE_ASYNC_LDS
    __builtin_amdgcn_s_wait_asynccnt(1);
#endif
}
__device__ __forceinline__ void wait_async_le0() {
#if USE_ASYNC_LDS
    __builtin_amdgcn_s_wait_asynccnt(0);
#endif
}

// ---------------------------------------------------------------------------
__global__ __launch_bounds__(256)
void nerf_march_kernel(const float* __restrict__ density,   // [N,S]
                       const float* __restrict__ shift,     // [S]
                       const float* __restrict__ rgb,       // [N,S,3]
                       const float* __restrict__ bg,        // [3]
                       const int*   __restrict__ interval_p,// scalar (python int)
                       float* __restrict__ out_rgb,         // [N,3]
                       float* __restrict__ out_w,           // [N,S]
                       float* __restrict__ out_ainv,        // [N]
                       int n_rays)
{
    __shared__ float s_shift[S_SAMPLES];
    __shared__ float s_dens[RAYS_PER_BLOCK][2][CHUNK];   // double buffer / wave

    const int tid  = threadIdx.x;
    const int lane = tid & 31;
    const int wave = tid >> 5;
    const int ray  = blockIdx.x * RAYS_PER_BLOCK + wave;

    // block-wide: stage shift[] once
    s_shift[tid] = shift[tid];
    __syncthreads();

    if (ray >= n_rays) return;

    const float interval = (float)interval_p[0];
    const float bg0 = bg[0], bg1 = bg[1], bg2 = bg[2];

    const float* __restrict__ dens_row = density + (size_t)ray * S_SAMPLES;
    const float* __restrict__ rgb_row  = rgb + (size_t)ray * S_SAMPLES * 3;
    float* __restrict__ w_row = out_w + (size_t)ray * S_SAMPLES;

    // prime the async pipeline: chunk 0 -> buffer 0
    stage_b32(dens_row + lane, &s_dens[wave][0][lane]);

    float carry = 1.0f;                 // running Π(1-alpha) across chunks
    float accr = 0.0f, accg = 0.0f, accb = 0.0f;

    #pragma unroll
    for (int j = 0; j < NCHUNK; ++j) {
        // prefetch next density chunk into the other buffer (ASYNCcnt path)
        if (j + 1 < NCHUNK)
            stage_b32(dens_row + (j + 1) * CHUNK + lane,
                      &s_dens[wave][(j + 1) & 1][lane]);
        // prefetch rgb two chunks ahead into cache (global_prefetch_b8)
        if (j + 2 < NCHUNK)
            __builtin_prefetch(rgb_row + (size_t)(j + 2) * CHUNK * 3 + lane * 3, 0, 0);

        // wait: all but the just-issued async op complete
        if (j + 1 < NCHUNK) wait_async_le1(); else wait_async_le0();

        const int s = j * CHUNK + lane;
        float d = s_dens[wave][j & 1][lane];

        // stable softplus, then alpha = -expm1(-softplus * interval)
        float x  = d + s_shift[s];
        float sp = fmaxf(x, 0.0f) + log1pf(expf(-fabsf(x)));
        float alpha = -expm1f(-sp * interval);
        float om = 1.0f - alpha;

        // wave32 inclusive prefix product of om
        float p = om;
        #pragma unroll
        for (int dlt = 1; dlt < 32; dlt <<= 1) {
            float y = __shfl_up(p, dlt, 32);
            p *= (lane >= dlt) ? y : 1.0f;
        }
        float chunk_total = __shfl(p, 31, 32);       // Π over this chunk
        float excl = __shfl_up(p, 1, 32);            // exclusive prefix
        if (lane == 0) excl = 1.0f;

        float T_excl = carry * excl;
        float w = T_excl * alpha;
        w_row[s] = w;                                // coalesced store

        // per-ray weighted rgb accumulation (not a GEMM: rgb depends on n)
        const float* c3 = rgb_row + (size_t)s * 3;
        float r = c3[0], g = c3[1], b = c3[2];
        accr = fmaf(w, r, accr);
        accg = fmaf(w, g, accg);
        accb = fmaf(w, b, accb);

        carry *= chunk_total;
    }

    // butterfly reduce the 3 channel accumulators across the wave
    #pragma unroll
    for (int m = 16; m >= 1; m >>= 1) {
        accr += __shfl_xor(accr, m, 32);
        accg += __shfl_xor(accg, m, 32);
        accb += __shfl_xor(accb, m, 32);
    }

    if (lane == 0) {
        out_rgb[(size_t)ray * 3 + 0] = accr + carry * bg0;
        out_rgb[(size_t)ray * 3 + 1] = accg + carry * bg1;
        out_rgb[(size_t)ray * 3 + 2] = accb + carry * bg2;
        out_ainv[ray] = carry;
    }
}

// ---------------------------------------------------------------------------
extern "C" void kernel_launch(void* const* d_in, const int* in_sizes, int n_in,
                              void* d_out, int out_size, void* d_ws, size_t ws_size,
                              hipStream_t stream) {
    const float* density  = (const float*)d_in[0];   // [N,S]
    const float* shift    = (const float*)d_in[1];   // [S]
    const float* rgb      = (const float*)d_in[2];   // [N,S,3]
    const float* bg       = (const float*)d_in[3];   // [3]
    const int*   interval = (const int*)d_in[4];     // scalar python int

    const int n_rays = in_sizes[0] / S_SAMPLES;      // 32768

    float* out      = (float*)d_out;
    float* out_rgb  = out;                                        // N*3
    float* out_w    = out + (size_t)n_rays * 3;                   // N*S
    float* out_ainv = out + (size_t)n_rays * 3 + (size_t)n_rays * S_SAMPLES;

    dim3 grid((n_rays + RAYS_PER_BLOCK - 1) / RAYS_PER_BLOCK);
    dim3 block(RAYS_PER_BLOCK * 32);
    nerf_march_kernel<<<grid, block, 0, stream>>>(
        density, shift, rgb, bg, interval, out_rgb, out_w, out_ainv, n_rays);
}